// BayesianGCN_59442347377126
// MI455X (gfx1250) — compile-verified
//
#include <hip/hip_runtime.h>

// Problem constants (match reference)
#define NN   50000
#define NE   800000
#define DIN_ 128
#define HH   128
#define CC   64

typedef __attribute__((ext_vector_type(2))) float v2f;
typedef __attribute__((ext_vector_type(8))) float v8f;

// ---------------------------------------------------------------------------
// Degree / normalization: deg[i] = 2 (double self loops) + in-degree(dst)
// ---------------------------------------------------------------------------
__global__ __launch_bounds__(256) void k_init_deg(float* __restrict__ deg) {
  int i = blockIdx.x * blockDim.x + threadIdx.x;
  if (i < NN) deg[i] = 2.0f;
}

__global__ __launch_bounds__(256) void k_deg_accum(const int* __restrict__ dst,
                                                   float* __restrict__ deg) {
  int e = blockIdx.x * blockDim.x + threadIdx.x;
  if (e < NE) unsafeAtomicAdd(&deg[dst[e]], 1.0f);  // native global_atomic_add_f32
}

__global__ __launch_bounds__(256) void k_dinv(const float* __restrict__ deg,
                                              float* __restrict__ dinv) {
  int i = blockIdx.x * blockDim.x + threadIdx.x;
  if (i < NN) dinv[i] = rsqrtf(deg[i]);  // deg >= 2 always (double self loops)
}

// Vectorized zero-fill: one b128 store per thread per iteration. n % 4 == 0.
__global__ __launch_bounds__(256) void k_zero4(float4* __restrict__ p, int n4) {
  int i = blockIdx.x * blockDim.x + threadIdx.x;
  int stride = gridDim.x * blockDim.x;
  const float4 z = {0.f, 0.f, 0.f, 0.f};
  for (; i < n4; i += stride) p[i] = z;
}

// ---------------------------------------------------------------------------
// Dense GEMM  D[NN x NCOL] = A[NN x K] * B[K x NCOL]  via V_WMMA_F32_16X16X4_F32
// One wave per 16x16 output tile; K-loop in steps of 4 (fully unrolled).
// A frag (16x4 f32, 2 VGPRs): lanes 0-15 -> K = k0+0,k0+1 ; lanes 16-31 -> K = k0+2,k0+3
// B frag (4x16 f32, 2 VGPRs): mirrored (lane&15 = N column)
// C/D (8 VGPRs): VGPR v -> M = v (lanes 0-15), M = v+8 (lanes 16-31)
// ---------------------------------------------------------------------------
template <int K, int NCOL>
__global__ __launch_bounds__(256) void k_gemm_wmma(const float* __restrict__ A,
                                                   const float* __restrict__ B,
                                                   float* __restrict__ D) {
  const int lane = threadIdx.x & 31;
  const int wave = threadIdx.x >> 5;
  const int tile = blockIdx.x * (blockDim.x >> 5) + wave;
  constexpr int colTiles = NCOL / 16;
  constexpr int rowTiles = NN / 16;  // 50000/16 = 3125 exact
  if (tile >= rowTiles * colTiles) return;  // whole-wave uniform exit (EXEC all-1 inside)
  const int rt = tile / colTiles;
  const int ct = tile - rt * colTiles;
  const int row0 = rt * 16, col0 = ct * 16;

  const int m  = lane & 15;
  const int kb = (lane >> 4) << 1;  // 0 for lanes 0-15, 2 for lanes 16-31

  const float* arow = A + (size_t)(row0 + m) * K;
  const float* bcol = B + col0 + m;

  v8f acc = {};
#pragma unroll
  for (int k0 = 0; k0 < K; k0 += 4) {
    const float2 av = *(const float2*)(arow + k0 + kb);  // 8B aligned (k0+kb even)
    v2f a; a.x = av.x; a.y = av.y;
    v2f b; b.x = bcol[(size_t)(k0 + kb) * NCOL];
           b.y = bcol[(size_t)(k0 + kb + 1) * NCOL];
    acc = __builtin_amdgcn_wmma_f32_16x16x4_f32(
        /*neg_a=*/false, a, /*neg_b=*/false, b,
        /*c_mod=*/(short)0, acc, /*reuse_a=*/false, /*reuse_b=*/false);
  }

  const int mhi = (lane >> 4) << 3;  // 0 or 8
  float* drow = D + col0 + m;
#pragma unroll
  for (int v = 0; v < 8; ++v)
    drow[(size_t)(row0 + mhi + v) * NCOL] = acc[v];
}

// ---------------------------------------------------------------------------
// Edge scatter: accum[dst] += dinv[src]*dinv[dst] * h[src]   (one wave per edge)
// Each lane owns F/32 consecutive floats -> one b128/b64 load per lane
// (wave covers F*4 contiguous bytes), then native f32 atomics into accum.
// Working set sits in the 192MB L2, so minimizing issue count is what matters.
// ---------------------------------------------------------------------------
template <int F>
__global__ __launch_bounds__(256) void k_scatter(const int* __restrict__ src,
                                                 const int* __restrict__ dst,
                                                 const float* __restrict__ dinv,
                                                 const float* __restrict__ h,
                                                 float* __restrict__ accum) {
  const int lane = threadIdx.x & 31;
  const int e = (int)((blockIdx.x * (size_t)blockDim.x + threadIdx.x) >> 5);
  if (e >= NE) return;  // whole-wave uniform
  const int s = src[e];
  const int d = dst[e];
  const float norm = dinv[s] * dinv[d];
  constexpr int PER = F / 32;  // 4 (layer1) or 2 (layer2)
  const float* hp = h + (size_t)s * F + lane * PER;
  float* ap = accum + (size_t)d * F + lane * PER;
  if constexpr (PER == 4) {
    const float4 v = *(const float4*)hp;   // global_load_b128
    unsafeAtomicAdd(&ap[0], norm * v.x);
    unsafeAtomicAdd(&ap[1], norm * v.y);
    unsafeAtomicAdd(&ap[2], norm * v.z);
    unsafeAtomicAdd(&ap[3], norm * v.w);
  } else {
    const float2 v = *(const float2*)hp;   // global_load_b64
    unsafeAtomicAdd(&ap[0], norm * v.x);
    unsafeAtomicAdd(&ap[1], norm * v.y);
  }
}

// ---------------------------------------------------------------------------
// Layer-1 epilogue (in place on accum): add 2*dinv^2 self-loop term + bias, ReLU
// ---------------------------------------------------------------------------
__global__ __launch_bounds__(256) void k_post1(float* __restrict__ accum,
                                               const float* __restrict__ h,
                                               const float* __restrict__ dinv,
                                               const float* __restrict__ b1) {
  int idx = blockIdx.x * blockDim.x + threadIdx.x;
  if (idx >= NN * HH) return;
  const int i = idx >> 7;        // / 128
  const int j = idx & (HH - 1);  // % 128
  const float di = dinv[i];
  const float v = accum[idx] + 2.0f * di * di * h[idx] + b1[j];
  accum[idx] = v > 0.0f ? v : 0.0f;
}

// ---------------------------------------------------------------------------
// Layer-2 epilogue + row-wise log_softmax over C=64 (one wave per node)
// ---------------------------------------------------------------------------
__global__ __launch_bounds__(256) void k_logsoftmax(const float* __restrict__ accum,
                                                    const float* __restrict__ g,
                                                    const float* __restrict__ dinv,
                                                    const float* __restrict__ b2,
                                                    float* __restrict__ out) {
  const int lane = threadIdx.x & 31;
  const int i = (int)((blockIdx.x * (size_t)blockDim.x + threadIdx.x) >> 5);
  if (i >= NN) return;
  const float di = dinv[i];
  const float sl = 2.0f * di * di;
  const size_t base = (size_t)i * CC;
  float v0 = accum[base + lane]      + sl * g[base + lane]      + b2[lane];
  float v1 = accum[base + lane + 32] + sl * g[base + lane + 32] + b2[lane + 32];
  float m = fmaxf(v0, v1);
#pragma unroll
  for (int off = 16; off > 0; off >>= 1) m = fmaxf(m, __shfl_xor(m, off, 32));
  float s = expf(v0 - m) + expf(v1 - m);
#pragma unroll
  for (int off = 16; off > 0; off >>= 1) s += __shfl_xor(s, off, 32);
  const float lse = m + logf(s);
  out[base + lane]      = v0 - lse;
  out[base + lane + 32] = v1 - lse;
}

// ---------------------------------------------------------------------------
// Orchestration
// ---------------------------------------------------------------------------
extern "C" void kernel_launch(void* const* d_in, const int* in_sizes, int n_in,
                              void* d_out, int out_size, void* d_ws, size_t ws_size,
                              hipStream_t stream) {
  const float* x  = (const float*)d_in[0];                 // [NN, DIN]
  const int*   ei = (const int*)d_in[1];                   // [2, NE] row-major
  const float* W1 = (const float*)d_in[2];                 // [DIN, HH]
  const float* b1 = (const float*)d_in[3];                 // [HH]
  const float* W2 = (const float*)d_in[4];                 // [HH, CC]
  const float* b2 = (const float*)d_in[5];                 // [CC]
  const int* src = ei;
  const int* dst = ei + NE;

  // Workspace carve-out (~77 MB)
  float* deg  = (float*)d_ws;          // NN
  float* dinv = deg  + NN;             // NN
  float* H1   = dinv + NN;             // NN*HH   (X @ W1)
  float* A1   = H1   + (size_t)NN*HH;  // NN*HH   (aggregated layer-1, ReLU'd in place)
  float* G2   = A1   + (size_t)NN*HH;  // NN*CC   (A1 @ W2)
  float* A2   = G2   + (size_t)NN*CC;  // NN*CC   (aggregated layer-2)
  float* out  = (float*)d_out;         // NN*CC log-probs

  const int T = 256;

  // 1) degrees / norms
  k_init_deg<<<(NN + T - 1) / T, T, 0, stream>>>(deg);
  k_deg_accum<<<(NE + T - 1) / T, T, 0, stream>>>(dst, deg);
  k_dinv<<<(NN + T - 1) / T, T, 0, stream>>>(deg, dinv);

  // 2) zero accumulators (d_ws is poisoned, not zeroed, by harness)
  k_zero4<<<2048, T, 0, stream>>>((float4*)A1, NN * HH / 4);
  k_zero4<<<2048, T, 0, stream>>>((float4*)A2, NN * CC / 4);

  // 3) Layer 1: H1 = X @ W1 (WMMA f32), scatter, epilogue(ReLU+bias+selfloops)
  {
    const int tiles = (NN / 16) * (HH / 16);
    k_gemm_wmma<DIN_, HH><<<(tiles + 7) / 8, T, 0, stream>>>(x, W1, H1);
  }
  k_scatter<HH><<<(NE + 7) / 8, T, 0, stream>>>(src, dst, dinv, H1, A1);
  k_post1<<<(NN * HH + T - 1) / T, T, 0, stream>>>(A1, H1, dinv, b1);

  // 4) Layer 2: G2 = A1 @ W2 (WMMA f32), scatter, epilogue + log_softmax
  {
    const int tiles = (NN / 16) * (CC / 16);
    k_gemm_wmma<HH, CC><<<(tiles + 7) / 8, T, 0, stream>>>(A1, W2, G2);
  }
  k_scatter<CC><<<(NE + 7) / 8, T, 0, stream>>>(src, dst, dinv, G2, A2);
  k_logsoftmax<<<(NN + 7) / 8, T, 0, stream>>>(A2, G2, dinv, b2, out);
}